// LinearExperts_21921513079446
// MI455X (gfx1250) — compile-verified
//
#include <hip/hip_runtime.h>

#define E_        8
#define H_        1024
#define D_        2048
#define STRIDE_   (D_ + 2)        // LDS row stride: 2050 % 64 != 0 -> bank-conflict-free
#define TOPK_     2
#define T_        1024            // B*S = 2*512
#define PAIRS_    (T_ * TOPK_)    // 2048 (token, k) routing pairs
#define MT_       32              // tokens per workgroup (two stacked WMMA M-tiles)
#define MAX_TILES_ (PAIRS_ / MT_) // 64 worst-case tiles per expert
#define ALPHA_    1.702f
#define LIMIT_    7.0f

typedef __attribute__((ext_vector_type(2))) float v2f;
typedef __attribute__((ext_vector_type(8))) float v8f;

// ---------------------------------------------------------------------------
// Kernel 1: deterministic routing. Lane e (< E_) scans all (t,k) pairs in
// order and builds a stable compacted list for expert e:
//   listTok  : source token id
//   listSlot : output slot = t*TOPK_+k  (unique per pair -> race-free scatter)
//   listW    : routing weight (duplicate expert in top-k => two pairs, which
//              reproduces the reference's mask-multiplicity semantics)
// Lists padded to a multiple of MT_ with weight-0 entries -> dummy slot.
// ---------------------------------------------------------------------------
__global__ void route_kernel(const int* __restrict__ idx,
                             const float* __restrict__ rw,
                             int* __restrict__ listTok,
                             int* __restrict__ listSlot,
                             float* __restrict__ listW,
                             int* __restrict__ counts) {
  const int e = threadIdx.x;
  if (e >= E_) return;
  int cnt = 0;
  for (int p = 0; p < PAIRS_; ++p) {
    const int t = p >> 1;  // TOPK_ == 2
    if (idx[p] == e) {
      listTok[e * PAIRS_ + cnt]  = t;
      listSlot[e * PAIRS_ + cnt] = p;
      listW[e * PAIRS_ + cnt]    = rw[t * E_ + e];
      ++cnt;
    }
  }
  counts[e] = cnt;
  const int padded = (cnt + MT_ - 1) & ~(MT_ - 1);
  for (int j = cnt; j < padded; ++j) {
    listTok[e * PAIRS_ + j]  = 0;       // valid row, result discarded
    listSlot[e * PAIRS_ + j] = PAIRS_;  // dummy slot
    listW[e * PAIRS_ + j]    = 0.0f;
  }
}

// ---------------------------------------------------------------------------
// Kernel 2: fused expert MLP for one 32-token tile of one expert.
// Register blocking: 2 M-tiles (32 tokens) x 2 adjacent N-tiles per wave, so
// every A fragment feeds 4 WMMAs and every B cache line feeds 2 N-tiles.
//   Stage 1: act[32, D] = (clip(up)+1) * glu(min(gate, LIMIT))  in ~256KB LDS
//            (row stride padded to 2050 floats -> conflict-free stage-2 reads)
//   Stage 2: y[32, H]   = act @ Wd + bd, scaled by routing weight, scattered
//            to per-pair slots in workspace.
// All GEMM work uses V_WMMA_F32_16X16X4_F32 (exact fp32).
//
// f32 WMMA operand layout (wave32):
//   A (16x4):  v0 = {lanes 0-15: (M=lane, K=k0),   lanes 16-31: (M=lane-16, K=k0+2)}
//              v1 = same M, K+1  -> per lane one float2 load of (K, K+1)
//   B (4x16):  v0 = {lanes 0-15: (K=k0, N=lane),   lanes 16-31: (K=k0+2, N=lane-16)}
//              v1 = same N, K+1  -> two b32 loads, D (or H) floats apart
//   C/D:       VGPR r = {lanes 0-15: M=r, N=lane; lanes 16-31: M=8+r, N=lane-16}
// ---------------------------------------------------------------------------
__global__ __launch_bounds__(256, 1)
void expert_kernel(const float* __restrict__ x,
                   const float* __restrict__ Wg, const float* __restrict__ bg,
                   const float* __restrict__ Wu, const float* __restrict__ bu,
                   const float* __restrict__ Wd, const float* __restrict__ bd,
                   const int* __restrict__ listTok,
                   const int* __restrict__ listSlot,
                   const float* __restrict__ listW,
                   const int* __restrict__ counts,
                   float* __restrict__ ysc) {
  __shared__ float act[MT_ * STRIDE_];   // ~256.25 KB of the WGP's 320 KB LDS
  __shared__ int   sTok[MT_];
  __shared__ int   sSlot[MT_];
  __shared__ float sWgt[MT_];

  const int e    = blockIdx.x / MAX_TILES_;
  const int tile = blockIdx.x % MAX_TILES_;
  const int cnt  = counts[e];
  const int padded = (cnt + MT_ - 1) & ~(MT_ - 1);
  if (tile * MT_ >= padded) return;   // block-uniform

  if (threadIdx.x < MT_) {
    const int g = e * PAIRS_ + tile * MT_ + threadIdx.x;
    sTok[threadIdx.x]  = listTok[g];
    sSlot[threadIdx.x] = listSlot[g];
    sWgt[threadIdx.x]  = listW[g];
  }
  __syncthreads();

  const int lane = threadIdx.x & 31;
  const int wave = threadIdx.x >> 5;   // 0..7
  const int half = lane >> 4;          // lane group (K offset 0 or 2)
  const int l16  = lane & 15;

  // ---------------- Stage 1: gate/up GEMMs + activation into LDS -----------
  const float* aRow0 = x + (size_t)sTok[l16]      * H_ + 2 * half;  // M-tile 0
  const float* aRow1 = x + (size_t)sTok[l16 + 16] * H_ + 2 * half;  // M-tile 1
  const float* wg = Wg + (size_t)e * H_ * D_;
  const float* wu = Wu + (size_t)e * H_ * D_;

  for (int it = 0; it < 8; ++it) {     // 128 N-tiles: 8 iters x 8 waves x 2 tiles
    const int n0 = (it * 16 + wave * 2) * 16;   // this wave owns cols n0..n0+31
    const float* bgP = wg + (size_t)(2 * half) * D_ + n0 + l16;
    const float* buP = wu + (size_t)(2 * half) * D_ + n0 + l16;
    v8f cg00 = {}, cg01 = {}, cg10 = {}, cg11 = {};   // [M-tile][N-tile]
    v8f cu00 = {}, cu01 = {}, cu10 = {}, cu11 = {};
#pragma unroll 2
    for (int k0 = 0; k0 < H_; k0 += 4) {
      v2f a0, a1;
      { const float2 t2 = *(const float2*)(aRow0 + k0); a0.x = t2.x; a0.y = t2.y; }
      { const float2 t2 = *(const float2*)(aRow1 + k0); a1.x = t2.x; a1.y = t2.y; }
      v2f bg0, bg1, bu0, bu1;
      bg0.x = bgP[(size_t)k0 * D_];
      bg0.y = bgP[(size_t)k0 * D_ + D_];
      bg1.x = bgP[(size_t)k0 * D_ + 16];
      bg1.y = bgP[(size_t)k0 * D_ + D_ + 16];
      bu0.x = buP[(size_t)k0 * D_];
      bu0.y = buP[(size_t)k0 * D_ + D_];
      bu1.x = buP[(size_t)k0 * D_ + 16];
      bu1.y = buP[(size_t)k0 * D_ + D_ + 16];
      cg00 = __builtin_amdgcn_wmma_f32_16x16x4_f32(false, a0, false, bg0, (short)0, cg00, false, false);
      cg10 = __builtin_amdgcn_wmma_f32_16x16x4_f32(false, a1, false, bg0, (short)0, cg10, false, false);
      cg01 = __builtin_amdgcn_wmma_f32_16x16x4_f32(false, a0, false, bg1, (short)0, cg01, false, false);
      cg11 = __builtin_amdgcn_wmma_f32_16x16x4_f32(false, a1, false, bg1, (short)0, cg11, false, false);
      cu00 = __builtin_amdgcn_wmma_f32_16x16x4_f32(false, a0, false, bu0, (short)0, cu00, false, false);
      cu10 = __builtin_amdgcn_wmma_f32_16x16x4_f32(false, a1, false, bu0, (short)0, cu10, false, false);
      cu01 = __builtin_amdgcn_wmma_f32_16x16x4_f32(false, a0, false, bu1, (short)0, cu01, false, false);
      cu11 = __builtin_amdgcn_wmma_f32_16x16x4_f32(false, a1, false, bu1, (short)0, cu11, false, false);
    }
#pragma unroll
    for (int nb = 0; nb < 2; ++nb) {
      const int col = n0 + nb * 16 + l16;
      const float biasG = bg[(size_t)e * D_ + col];
      const float biasU = bu[(size_t)e * D_ + col];
      const v8f& cgA = nb ? cg01 : cg00;
      const v8f& cgB = nb ? cg11 : cg10;
      const v8f& cuA = nb ? cu01 : cu00;
      const v8f& cuB = nb ? cu11 : cu10;
#pragma unroll
      for (int r = 0; r < 8; ++r) {
        const int m0 = r + 8 * half;       // rows 0..15
        const int m1 = m0 + 16;            // rows 16..31
        float g0 = cgA[r] + biasG, g1 = cgB[r] + biasG;
        float u0 = cuA[r] + biasU, u1 = cuB[r] + biasU;
        g0 = fminf(g0, LIMIT_);
        g1 = fminf(g1, LIMIT_);
        u0 = fminf(fmaxf(u0, -LIMIT_), LIMIT_);
        u1 = fminf(fmaxf(u1, -LIMIT_), LIMIT_);
        const float glu0 = g0 / (1.0f + __expf(-ALPHA_ * g0));
        const float glu1 = g1 / (1.0f + __expf(-ALPHA_ * g1));
        act[m0 * STRIDE_ + col] = (u0 + 1.0f) * glu0;
        act[m1 * STRIDE_ + col] = (u1 + 1.0f) * glu1;
      }
    }
  }
  __syncthreads();

  // ---------------- Stage 2: down GEMM + weighted scatter -------------------
  const float* wd = Wd + (size_t)e * D_ * H_;
  const float* aL0 = act + l16 * STRIDE_ + 2 * half;          // LDS A-tile 0
  const float* aL1 = act + (l16 + 16) * STRIDE_ + 2 * half;   // LDS A-tile 1

  for (int it = 0; it < 4; ++it) {     // 64 N-tiles: 4 iters x 8 waves x 2 tiles
    const int n0 = (it * 16 + wave * 2) * 16;
    const float* bP = wd + (size_t)(2 * half) * H_ + n0 + l16;
    v8f c00 = {}, c01 = {}, c10 = {}, c11 = {};
#pragma unroll 2
    for (int k0 = 0; k0 < D_; k0 += 4) {
      v2f a0, a1;
      { const float2 t2 = *(const float2*)(aL0 + k0); a0.x = t2.x; a0.y = t2.y; }
      { const float2 t2 = *(const float2*)(aL1 + k0); a1.x = t2.x; a1.y = t2.y; }
      v2f b0, b1;
      b0.x = bP[(size_t)k0 * H_];
      b0.y = bP[(size_t)k0 * H_ + H_];
      b1.x = bP[(size_t)k0 * H_ + 16];
      b1.y = bP[(size_t)k0 * H_ + H_ + 16];
      c00 = __builtin_amdgcn_wmma_f32_16x16x4_f32(false, a0, false, b0, (short)0, c00, false, false);
      c10 = __builtin_amdgcn_wmma_f32_16x16x4_f32(false, a1, false, b0, (short)0, c10, false, false);
      c01 = __builtin_amdgcn_wmma_f32_16x16x4_f32(false, a0, false, b1, (short)0, c01, false, false);
      c11 = __builtin_amdgcn_wmma_f32_16x16x4_f32(false, a1, false, b1, (short)0, c11, false, false);
    }
#pragma unroll
    for (int nb = 0; nb < 2; ++nb) {
      const int col = n0 + nb * 16 + l16;
      const float bias = bd[(size_t)e * H_ + col];
      const v8f& cA = nb ? c01 : c00;
      const v8f& cB = nb ? c11 : c10;
#pragma unroll
      for (int r = 0; r < 8; ++r) {
        const int m0 = r + 8 * half;
        const int m1 = m0 + 16;
        const float v0 = (cA[r] + bias) * sWgt[m0];
        const float v1 = (cB[r] + bias) * sWgt[m1];
        ysc[(size_t)sSlot[m0] * H_ + col] = v0;   // race-free per-pair slot
        ysc[(size_t)sSlot[m1] * H_ + col] = v1;
      }
    }
  }
}

// ---------------------------------------------------------------------------
// Kernel 3: combine the two top-k slots of each token (deterministic).
// ---------------------------------------------------------------------------
__global__ void combine_kernel(const float* __restrict__ ysc,
                               float* __restrict__ out) {
  const int i = blockIdx.x * blockDim.x + threadIdx.x;  // [0, T_*H_)
  const int t = i >> 10;           // / H_
  const int h = i & (H_ - 1);
  out[i] = ysc[(size_t)(2 * t) * H_ + h] + ysc[(size_t)(2 * t + 1) * H_ + h];
}

extern "C" void kernel_launch(void* const* d_in, const int* in_sizes, int n_in,
                              void* d_out, int out_size, void* d_ws, size_t ws_size,
                              hipStream_t stream) {
  const float* x   = (const float*)d_in[0];  // hidden_states [T, H]
  const int*   idx = (const int*)d_in[1];    // router_indices [T, K]
  const float* rw  = (const float*)d_in[2];  // routing_weights [T, E]
  const float* Wg  = (const float*)d_in[3];
  const float* bg  = (const float*)d_in[4];
  const float* Wu  = (const float*)d_in[5];
  const float* bu  = (const float*)d_in[6];
  const float* Wd  = (const float*)d_in[7];
  const float* bd  = (const float*)d_in[8];
  float* out = (float*)d_out;

  // Workspace carve-up (~8.6 MB total)
  char* ws = (char*)d_ws;
  int*   listTok  = (int*)ws;    ws += (size_t)E_ * PAIRS_ * sizeof(int);
  int*   listSlot = (int*)ws;    ws += (size_t)E_ * PAIRS_ * sizeof(int);
  float* listW    = (float*)ws;  ws += (size_t)E_ * PAIRS_ * sizeof(float);
  int*   counts   = (int*)ws;    ws += 256;
  float* ysc      = (float*)ws;  // (PAIRS_+1) * H_ floats (last row = dummy)

  route_kernel<<<1, 32, 0, stream>>>(idx, rw, listTok, listSlot, listW, counts);
  expert_kernel<<<E_ * MAX_TILES_, 256, 0, stream>>>(
      x, Wg, bg, Wu, bu, Wd, bd, listTok, listSlot, listW, counts, ysc);
  combine_kernel<<<(T_ * H_) / 256, 256, 0, stream>>>(ysc, out);
}